// MRNormer_81423989997775
// MI455X (gfx1250) — compile-verified
//
#include <hip/hip_runtime.h>
#include <hip/hip_bf16.h>
#include <math.h>

typedef __attribute__((ext_vector_type(2))) float v2f;
typedef __attribute__((ext_vector_type(4))) float v4f;
typedef __attribute__((ext_vector_type(8))) float v8f;
typedef __attribute__((ext_vector_type(4))) int   v4i;

#define N_NODE 20000
#define HIDD   128
#define NE     320000
#define NNF    (N_NODE * HIDD)   // 2,560,000 floats per node tensor
#define LDSPAD 132               // padded row stride (floats) -> conflict-free ds_load_b64

#if defined(__has_builtin)
#if __has_builtin(__builtin_amdgcn_global_load_async_to_lds_b128)
#define HAVE_ASYNC_LDS 1
#endif
#endif

__device__ __forceinline__ void wait_async0() {
#if defined(HAVE_ASYNC_LDS)
#if __has_builtin(__builtin_amdgcn_s_wait_asynccnt)
    __builtin_amdgcn_s_wait_asynccnt(0);
#else
    asm volatile("s_wait_asynccnt 0" ::: "memory");
#endif
#endif
}

// ---------- monotonic float <-> uint for atomicMax over signed floats ----------
__device__ __forceinline__ unsigned fflip(float f) {
    unsigned u = __float_as_uint(f);
    return (u & 0x80000000u) ? ~u : (u | 0x80000000u);
}
__device__ __forceinline__ float funflip(unsigned u) {
    return (u & 0x80000000u) ? __uint_as_float(u & 0x7FFFFFFFu)
                             : __uint_as_float(~u);
}

// ---------------------------------------------------------------------------
// Repack W[128x128] (k-major) -> WT[128x128] (n-major) so B fragments are
// contiguous 8-byte loads per lane.
// ---------------------------------------------------------------------------
__global__ __launch_bounds__(256) void transpose_w(
    const float* __restrict__ W, float* __restrict__ WT)
{
    int t = blockIdx.x * blockDim.x + threadIdx.x;   // 16384 threads
    int k = t >> 7, n = t & 127;
    WT[(size_t)n * HIDD + k] = W[t];
}

// ---------------------------------------------------------------------------
// GEMM: C[n x 128] = A[n x 128] @ W[128 x 128] + bias  (ACC -> += into C)
// Block = 256 thr = 8 waves; block covers a 16-row strip, wave w covers cols
// [16w,16w+16).  A strip (16x128, 8KB) is staged once into LDS via the async
// global->LDS path (ASYNCcnt + s_wait_asynccnt), then each wave feeds
// V_WMMA_F32_16X16X4_F32 from ds_load_b64 (A) + contiguous global b64 (B from
// repacked WT).  No divergence anywhere -> EXEC all ones for WMMA.
// ---------------------------------------------------------------------------
template<int ACC>
__global__ __launch_bounds__(256) void gemm_n128_wmma(
    const float* __restrict__ A, const float* __restrict__ WT,
    const float* __restrict__ bias, float* __restrict__ C)
{
    __shared__ float As[16 * LDSPAD];

    const int tid     = threadIdx.x;
    const int wave    = tid >> 5;
    const int lane    = tid & 31;
    const int rowBase = blockIdx.x << 4;
    const int colBase = wave << 4;
    const int mrow    = lane & 15;          // M row (and N col) for this lane
    const int khalf   = (lane >> 4) << 1;   // K sub-offset: 0 or 2

    // ---- stage A strip (16x128 f32) into LDS: 512 float4 chunks ----
    const float* Ablk = A + (size_t)rowBase * HIDD;
    #pragma unroll
    for (int c = tid; c < 512; c += 256) {           // uniform trip count (2)
        int row = c >> 5;                            // 0..15
        int ci  = c & 31;                            // float4 within row
        const float* g = Ablk + row * HIDD + ci * 4;
        float*       l = &As[row * LDSPAD + ci * 4];
#if defined(HAVE_ASYNC_LDS)
        __builtin_amdgcn_global_load_async_to_lds_b128(
            (__attribute__((address_space(1))) v4i*)g,
            (__attribute__((address_space(3))) v4i*)l, 0, 0);
#else
        *(v4f*)l = *(const v4f*)g;
#endif
    }
    wait_async0();
    __syncthreads();

    const float* Arow  = &As[mrow * LDSPAD];                 // LDS, bank-spread
    const float* WTcol = WT + (size_t)(colBase + mrow) * HIDD;

    v8f acc = {0.f,0.f,0.f,0.f,0.f,0.f,0.f,0.f};
    #pragma unroll 4
    for (int k = 0; k < HIDD; k += 4) {
        v2f a = *(const v2f*)(Arow  + k + khalf);    // ds_load_b64
        v2f b = *(const v2f*)(WTcol + k + khalf);    // global b64, contiguous
        acc = __builtin_amdgcn_wmma_f32_16x16x4_f32(
                  false, a, false, b, (short)0, acc, false, false);
    }

    const float bv     = bias[colBase + mrow];
    const int   rowOff = (lane >> 4) << 3;           // lanes 16-31: M = r+8
    #pragma unroll
    for (int r = 0; r < 8; ++r) {
        size_t idx = (size_t)(rowBase + rowOff + r) * HIDD + colBase + mrow;
        float  val = acc[r] + bv;
        if (ACC) val += C[idx];
        C[idx] = val;
    }
}

// ---------------------------------------------------------------------------
// Edge kernels: one thread per (edge, head).  HD = 16, NH = 8.
// ---------------------------------------------------------------------------
__global__ void edge_score_max(
    const float* __restrict__ K, const float* __restrict__ Q,
    const float* __restrict__ hw, const int* __restrict__ src,
    const int* __restrict__ dst, float* __restrict__ score,
    unsigned* __restrict__ segmax)
{
    int t = blockIdx.x * blockDim.x + threadIdx.x;
    if (t >= NE * 8) return;
    int e = t >> 3, h = t & 7;
    int s = src[e], d = dst[e];
    const float* kp = K  + (size_t)s * HIDD + h * 16;
    const float* qp = Q  + (size_t)d * HIDD + h * 16;
    const float* hp = hw + h * 16;
    float acc = 0.f;
    #pragma unroll
    for (int i = 0; i < 16; ++i) acc = fmaf(kp[i] * qp[i], hp[i], acc);
    acc *= 0.25f;                       // 1/sqrt(HD)
    score[t] = acc;
    atomicMax(&segmax[(size_t)d * 8 + h], fflip(acc));
}

__global__ void edge_exp_sum(
    const int* __restrict__ dst, const unsigned* __restrict__ segmax,
    float* __restrict__ score, float* __restrict__ segsum)
{
    int t = blockIdx.x * blockDim.x + threadIdx.x;
    if (t >= NE * 8) return;
    int e = t >> 3, h = t & 7;
    int d = dst[e];
    float m  = funflip(segmax[(size_t)d * 8 + h]);
    float ex = expf(score[t] - m);
    score[t] = ex;
    atomicAdd(&segsum[(size_t)d * 8 + h], ex);
}

__global__ void edge_aggregate(
    const float* __restrict__ V, const int* __restrict__ src,
    const int* __restrict__ dst, const float* __restrict__ score,
    const float* __restrict__ segsum, float* __restrict__ agg)
{
    int t = blockIdx.x * blockDim.x + threadIdx.x;
    if (t >= NE * 8) return;
    int e = t >> 3, h = t & 7;
    int s = src[e], d = dst[e];
    float alpha = score[t] / (segsum[(size_t)d * 8 + h] + 1e-16f);
    const float* vp = V   + (size_t)s * HIDD + h * 16;
    float*       ap = agg + (size_t)d * HIDD + h * 16;
    #pragma unroll
    for (int i = 0; i < 16; ++i) atomicAdd(&ap[i], alpha * vp[i]);
}

// ---------------------------------------------------------------------------
__global__ void fill_f32(float* p, float v, int n) {
    int t = blockIdx.x * blockDim.x + threadIdx.x;
    if (t < n) p[t] = v;
}
__global__ void fill_u32(unsigned* p, unsigned v, int n) {
    int t = blockIdx.x * blockDim.x + threadIdx.x;
    if (t < n) p[t] = v;
}
__global__ void elu_kernel(const float* __restrict__ in, float* __restrict__ out, int n) {
    int t = blockIdx.x * blockDim.x + threadIdx.x;
    if (t < n) { float x = in[t]; out[t] = (x > 0.f) ? x : expm1f(x); }
}

// ---------------------------------------------------------------------------
extern "C" void kernel_launch(void* const* d_in, const int* in_sizes, int n_in,
                              void* d_out, int out_size, void* d_ws, size_t ws_size,
                              hipStream_t stream)
{
    (void)in_sizes; (void)n_in; (void)out_size; (void)ws_size;

    const float* x_user = (const float*)d_in[0];
    const float* x_item = (const float*)d_in[1];
    const int*   e_ui   = (const int*)d_in[2];   // [2,E]: src(users), dst(items)
    const int*   e_iu   = (const int*)d_in[3];   // [2,E]: src(items), dst(users)

    // flattened params, setup_inputs() dict order (depth-first):
    //  4,5 in_user(W,b)   6,7 in_item   8,9 out_user  10,11 out_item
    //  rel_ui base 12 / rel_iu base 21 / rel_uu base 30 / rel_ii base 39
    //  rel layout: kW,kb,qW,qb,vW,vb,aW,ab,hw
    const float* P[48];
    for (int i = 4; i < 48; ++i) P[i] = (const float*)d_in[i];

    float* ws      = (float*)d_ws;
    float* H_USER  = ws;
    float* H_ITEM  = ws + (size_t)1 * NNF;
    float* M_USER  = ws + (size_t)2 * NNF;
    float* M_ITEM  = ws + (size_t)3 * NNF;
    float* KB      = ws + (size_t)4 * NNF;
    float* QB      = ws + (size_t)5 * NNF;
    float* VB      = ws + (size_t)6 * NNF;
    float* AGG     = ws + (size_t)7 * NNF;
    float* SC      = ws + (size_t)8 * NNF;             // E*8 == NNF floats
    unsigned* SEGMAX = (unsigned*)(ws + (size_t)9 * NNF);
    float*    SEGSUM = ws + (size_t)9 * NNF + (size_t)N_NODE * 8;
    float*    WT     = ws + (size_t)9 * NNF + (size_t)2 * N_NODE * 8;  // 16384 f

    const dim3 blk(256);
    const dim3 gGemm(N_NODE / 16);                      // 1250 row strips
    const dim3 gEdge((NE * 8 + 255) / 256);             // 10000
    const dim3 gNode((NNF + 255) / 256);
    const dim3 gSeg((N_NODE * 8 + 255) / 256);
    const dim3 gTr(HIDD * HIDD / 256);                  // 64

    // helper: repack W then run GEMM (stream-ordered, WT slot is reused)
    auto GEMM0 = [&](const float* A, const float* W, const float* b, float* C) {
        transpose_w<<<gTr, blk, 0, stream>>>(W, WT);
        gemm_n128_wmma<0><<<gGemm, blk, 0, stream>>>(A, WT, b, C);
    };
    auto GEMM1 = [&](const float* A, const float* W, const float* b, float* C) {
        transpose_w<<<gTr, blk, 0, stream>>>(W, WT);
        gemm_n128_wmma<1><<<gGemm, blk, 0, stream>>>(A, WT, b, C);
    };

    // ---- input projections ----
    GEMM0(x_user, P[4], P[5], H_USER);
    GEMM0(x_item, P[6], P[7], H_ITEM);

    // ---- m_user: rel_iu (items -> users), then rel_uu self-loop ----
    {
        const int *src = e_iu, *dst = e_iu + NE;        // base 21
        GEMM0(H_ITEM, P[21], P[22], KB);
        GEMM0(H_USER, P[23], P[24], QB);
        GEMM0(H_ITEM, P[25], P[26], VB);
        fill_u32<<<gSeg,  blk, 0, stream>>>(SEGMAX, 0x007FFFFFu, N_NODE * 8); // fflip(-inf)
        fill_f32<<<gSeg,  blk, 0, stream>>>(SEGSUM, 0.f, N_NODE * 8);
        fill_f32<<<gNode, blk, 0, stream>>>(AGG, 0.f, NNF);
        edge_score_max<<<gEdge, blk, 0, stream>>>(KB, QB, P[29], src, dst, SC, SEGMAX);
        edge_exp_sum  <<<gEdge, blk, 0, stream>>>(dst, SEGMAX, SC, SEGSUM);
        edge_aggregate<<<gEdge, blk, 0, stream>>>(VB, src, dst, SC, SEGSUM, AGG);
        GEMM0(AGG, P[27], P[28], M_USER);
    }
    // rel_uu: one self edge per node -> alpha==1 -> Lin_a(Lin_v(h_user))
    GEMM0(H_USER, P[34], P[35], VB);
    GEMM1(VB,     P[36], P[37], M_USER);

    // ---- m_item: rel_ui (users -> items), then rel_ii self-loop ----
    {
        const int *src = e_ui, *dst = e_ui + NE;        // base 12
        GEMM0(H_USER, P[12], P[13], KB);
        GEMM0(H_ITEM, P[14], P[15], QB);
        GEMM0(H_USER, P[16], P[17], VB);
        fill_u32<<<gSeg,  blk, 0, stream>>>(SEGMAX, 0x007FFFFFu, N_NODE * 8);
        fill_f32<<<gSeg,  blk, 0, stream>>>(SEGSUM, 0.f, N_NODE * 8);
        fill_f32<<<gNode, blk, 0, stream>>>(AGG, 0.f, NNF);
        edge_score_max<<<gEdge, blk, 0, stream>>>(KB, QB, P[20], src, dst, SC, SEGMAX);
        edge_exp_sum  <<<gEdge, blk, 0, stream>>>(dst, SEGMAX, SC, SEGSUM);
        edge_aggregate<<<gEdge, blk, 0, stream>>>(VB, src, dst, SC, SEGSUM, AGG);
        GEMM0(AGG, P[18], P[19], M_ITEM);
    }
    // rel_ii self-loop
    GEMM0(H_ITEM, P[43], P[44], VB);
    GEMM1(VB,     P[45], P[46], M_ITEM);

    // ---- output projections: out = Lin(elu(m)) ----
    float* out_user = (float*)d_out;
    float* out_item = (float*)d_out + NNF;
    elu_kernel<<<gNode, blk, 0, stream>>>(M_USER, KB, NNF);
    GEMM0(KB, P[8],  P[9],  out_user);
    elu_kernel<<<gNode, blk, 0, stream>>>(M_ITEM, KB, NNF);
    GEMM0(KB, P[10], P[11], out_item);
}